// MultiHeadAttention_83245056131083
// MI455X (gfx1250) — compile-verified
//
#include <hip/hip_runtime.h>
#include <hip/hip_bf16.h>

// ---------------------------------------------------------------------------
// Shapes (compile-time constants from the reference)
// ---------------------------------------------------------------------------
#define B 8
#define L 256
#define D 256
#define H 8
#define DH 32
#define NEGF (-4294967295.0f)          // -(2^32)+1
#define INV_SQRT_DH 0.17677669529663687f

typedef float  v2f  __attribute__((ext_vector_type(2)));
typedef float  v4f  __attribute__((ext_vector_type(4)));
typedef float  v8f  __attribute__((ext_vector_type(8)));

// ---------------------------------------------------------------------------
// Kernel 0: padding masks.  qmask[b*L+l] = (sum_d queries[b,l,d] != 0)
//           kmask likewise for keys.  One 256-thread block per row.
// ---------------------------------------------------------------------------
__global__ void mask_kernel(const float* __restrict__ queries,
                            const float* __restrict__ keys,
                            float* __restrict__ qmask,
                            float* __restrict__ kmask) {
    const int row = blockIdx.x;          // 0 .. B*L-1
    const int t   = threadIdx.x;         // 0 .. 255
    __shared__ float sq[D];
    __shared__ float sk[D];
    sq[t] = queries[row * D + t];
    sk[t] = keys[row * D + t];
    __syncthreads();
    #pragma unroll
    for (int s = 128; s > 0; s >>= 1) {
        if (t < s) { sq[t] += sq[t + s]; sk[t] += sk[t + s]; }
        __syncthreads();
    }
    if (t == 0) {
        qmask[row] = (sq[0] != 0.0f) ? 1.0f : 0.0f;
        kmask[row] = (sk[0] != 0.0f) ? 1.0f : 0.0f;
    }
}

// ---------------------------------------------------------------------------
// Kernel 1: fused Q/K/V projection GEMM via V_WMMA_F32_16X16X4_F32.
// X[2048,256] @ W[256,256] + bias, for 3 weight sets.
// One wave computes one 16x16 output tile; 8 waves (256 thr) per block.
// Tiles: (2048/16)*(256/16)=2048 per matrix, x3 = 6144 waves = 768 blocks.
//
// fp32 WMMA layouts (ISA 7.12.2):
//   A 16x4 : lanes 0-15 M=0..15, v0=K0 v1=K1 ; lanes 16-31 v0=K2 v1=K3
//   B 4x16 : lanes 0-15 N=0..15 K=0/1 in v0/v1 ; lanes 16-31 K=2/3
//   C/D    : vgpr r, lanes 0-15 -> M=r, lanes 16-31 -> M=r+8, N=lane%16
// ---------------------------------------------------------------------------
__global__ void proj_kernel(const float* __restrict__ queries,
                            const float* __restrict__ keys,
                            const float* __restrict__ Wq, const float* __restrict__ bq,
                            const float* __restrict__ Wk, const float* __restrict__ bk,
                            const float* __restrict__ Wv, const float* __restrict__ bv,
                            float* __restrict__ Qp,
                            float* __restrict__ Kp,
                            float* __restrict__ Vp) {
    const int wave = threadIdx.x >> 5;
    const int lane = threadIdx.x & 31;
    const int gw   = blockIdx.x * 8 + wave;      // 0 .. 6143
    const int mat  = gw / 2048;                  // 0=Q 1=K 2=V
    const int tile = gw % 2048;
    const int m0   = (tile >> 4) << 4;           // row-tile * 16
    const int n0   = (tile & 15) << 4;           // col-tile * 16

    const float* X    = (mat == 0) ? queries : keys;
    const float* W    = (mat == 0) ? Wq : (mat == 1) ? Wk : Wv;
    const float* bias = (mat == 0) ? bq : (mat == 1) ? bk : bv;
    float*       Out  = (mat == 0) ? Qp : (mat == 1) ? Kp : Vp;

    const int half = lane >> 4;                  // 0: K pair {0,1}, 1: {2,3}
    const int mr   = m0 + (lane & 15);
    const int nc   = n0 + (lane & 15);

    v8f acc = {};
    #pragma unroll 4
    for (int kk = 0; kk < D; kk += 4) {
        const int ka = kk + half * 2;
        v2f a = *(const v2f*)(X + mr * D + ka);          // A[m, ka], A[m, ka+1]
        v2f b;
        b.x = W[ka * D + nc];                            // B[ka,   n]
        b.y = W[(ka + 1) * D + nc];                      // B[ka+1, n]
        acc = __builtin_amdgcn_wmma_f32_16x16x4_f32(
            /*neg_a=*/false, a, /*neg_b=*/false, b,
            /*c_mod=*/(short)0, acc, /*reuse_a=*/false, /*reuse_b=*/false);
    }

    const float bb = bias[nc];
    #pragma unroll
    for (int r = 0; r < 8; ++r) {
        Out[(m0 + r + half * 8) * D + nc] = acc[r] + bb;
    }
}

// ---------------------------------------------------------------------------
// Kernel 2: attention for one (b, q) pair per 256-thread block.
//  Phase A: projected Q row -> LDS
//  Phase B: thread t = key k computes 8 per-head scores, fusing Kp + posK + Tk
//           (Tk streamed nontemporally; causal skip k>q)
//  Phase C: wave w softmaxes head w (wave32 shfl reduction)
//  Phase D: thread t = channel d accumulates attn*(Vp + posV + Tv), coalesced
// ---------------------------------------------------------------------------
__global__ void attn_kernel(const float* __restrict__ queries,
                            const float* __restrict__ apk,
                            const float* __restrict__ apv,
                            const float* __restrict__ tmk,
                            const float* __restrict__ tmv,
                            const float* __restrict__ Qp,
                            const float* __restrict__ Kp,
                            const float* __restrict__ Vp,
                            const float* __restrict__ qmask,
                            const float* __restrict__ kmask,
                            float* __restrict__ out) {
    const int q = blockIdx.x;           // 0..L-1
    const int b = blockIdx.y;           // 0..B-1
    const int t = threadIdx.x;          // 0..255
    const int bq = b * L + q;

    __shared__ float qrow[D];
    __shared__ float sc[H][L];          // scores, then attention weights

    // ---- Phase A: projected query row into LDS --------------------------
    qrow[t] = Qp[(size_t)bq * D + t];
    __syncthreads();

    // ---- Phase B: scores for key k = t ----------------------------------
    {
        const int k = t;
        float s[H];
        #pragma unroll
        for (int h = 0; h < H; ++h) s[h] = 0.0f;

        const bool masked = (k > q) || (kmask[b * L + k] == 0.0f);
        if (k <= q) {
            const v4f* kp = (const v4f*)(Kp  + (size_t)(b * L + k) * D);
            const v4f* pk = (const v4f*)(apk + (size_t)(b * L + k) * D);
            const v4f* tk = (const v4f*)(tmk + ((size_t)bq * L + k) * D);
            #pragma unroll 2
            for (int d4 = 0; d4 < D / 4; ++d4) {
                const int h = (d4 * 4) >> 5;
                v4f kv = kp[d4];
                v4f pv = pk[d4];
                v4f tv = __builtin_nontemporal_load(tk + d4);   // stream, NT
                #pragma unroll
                for (int j = 0; j < 4; ++j)
                    s[h] += qrow[d4 * 4 + j] * (kv[j] + pv[j] + tv[j]);
            }
        }
        #pragma unroll
        for (int h = 0; h < H; ++h)
            sc[h][k] = masked ? NEGF : s[h] * INV_SQRT_DH;
    }
    __syncthreads();

    // ---- Phase C: softmax, one wave32 per head --------------------------
    {
        const int h    = t >> 5;        // wave id == head
        const int lane = t & 31;
        float m = -3.4e38f;
        #pragma unroll
        for (int k = lane; k < L; k += 32) m = fmaxf(m, sc[h][k]);
        #pragma unroll
        for (int off = 16; off > 0; off >>= 1)
            m = fmaxf(m, __shfl_xor(m, off, 32));

        float sum = 0.0f;
        #pragma unroll
        for (int k = lane; k < L; k += 32) {
            float e = __expf(sc[h][k] - m);
            sc[h][k] = e;
            sum += e;
        }
        #pragma unroll
        for (int off = 16; off > 0; off >>= 1)
            sum += __shfl_xor(sum, off, 32);

        const float inv = qmask[bq] / sum;     // query-mask folded in
        #pragma unroll
        for (int k = lane; k < L; k += 32) sc[h][k] *= inv;
    }
    __syncthreads();

    // ---- Phase D: output channel d = t, coalesced row sweep -------------
    {
        const int h = t >> 5;
        float acc = 0.0f;
        const float* vrow = Vp  + (size_t)(b * L) * D + t;
        const float* prow = apv + (size_t)(b * L) * D + t;
        const float* trow = tmv + (size_t)bq * L * D + t;
        for (int k = 0; k <= q; ++k) {          // causal skip
            const float w  = sc[h][k];
            const float tv = __builtin_nontemporal_load(trow + (size_t)k * D);
            acc += w * (vrow[(size_t)k * D] + prow[(size_t)k * D] + tv);
        }
        out[(size_t)bq * D + t] = acc + queries[(size_t)bq * D + t];
    }
}

// ---------------------------------------------------------------------------
// Host launcher
// ---------------------------------------------------------------------------
extern "C" void kernel_launch(void* const* d_in, const int* in_sizes, int n_in,
                              void* d_out, int out_size, void* d_ws, size_t ws_size,
                              hipStream_t stream) {
    const float* queries = (const float*)d_in[0];
    const float* keys    = (const float*)d_in[1];
    const float* tmk     = (const float*)d_in[2];
    const float* tmv     = (const float*)d_in[3];
    const float* apk     = (const float*)d_in[4];
    const float* apv     = (const float*)d_in[5];
    const float* Wq      = (const float*)d_in[6];
    const float* bq      = (const float*)d_in[7];
    const float* Wk      = (const float*)d_in[8];
    const float* bk      = (const float*)d_in[9];
    const float* Wv      = (const float*)d_in[10];
    const float* bv      = (const float*)d_in[11];
    float* out = (float*)d_out;

    float* ws    = (float*)d_ws;
    const size_t NBL = (size_t)B * L * D;       // 524288
    float* Qp    = ws;
    float* Kp    = ws + NBL;
    float* Vp    = ws + 2 * NBL;
    float* qmask = ws + 3 * NBL;
    float* kmask = qmask + (size_t)B * L;

    // masks: one block per (b,l) row
    mask_kernel<<<B * L, 256, 0, stream>>>(queries, keys, qmask, kmask);

    // projections: 6144 waves (2048 tiles x 3 matrices), 8 waves per block
    proj_kernel<<<768, 256, 0, stream>>>(queries, keys, Wq, bq, Wk, bk, Wv, bv,
                                         Qp, Kp, Vp);

    // attention: one block per (b, q)
    attn_kernel<<<dim3(L, B), 256, 0, stream>>>(queries, apk, apv, tmk, tmv,
                                                Qp, Kp, Vp, qmask, kmask, out);
}